// Reconstruct_24352464569114
// MI455X (gfx1250) — compile-verified
//
#include <hip/hip_runtime.h>
#include <hip/hip_bf16.h>

typedef __attribute__((ext_vector_type(2))) float v2f;
typedef __attribute__((ext_vector_type(8))) float v8f;

#define Hc 384
#define Wc 1280
#define Cc 3
#define Fc 7
#define Kc 49      // number of filters == filter taps (7*7)
#define Pc 3
#define TILE_COLS 134   // 128 pixels + 2*3 halo
#define TILE_PITCH 136  // padded to dodge bank conflicts

__device__ __forceinline__ int reflect_idx(int i, int n) {
    // jnp.pad 'reflect' (no edge repeat); max overhang is 3 so one pass suffices
    if (i < 0) i = -i;
    if (i >= n) i = 2 * n - 2 - i;
    return i;
}

__global__ void __launch_bounds__(256)
reconstruct_wmma_kernel(const float* __restrict__ img,    // (3,H,W)
                        const float* __restrict__ offx,   // (H,W)
                        const float* __restrict__ offy,   // (H,W)
                        const float* __restrict__ filt,   // (49,1,7,7) -> flat (49,49)
                        float* __restrict__ out) {        // (3,H,W)
    __shared__ float ldsX[Fc * TILE_PITCH];
    __shared__ float ldsY[Fc * TILE_PITCH];
    __shared__ float wLds[Kc * Kc];

    const int blocksPerRow = Wc / 128;           // 10
    const int h  = blockIdx.x / blocksPerRow;
    const int w0 = (blockIdx.x % blocksPerRow) * 128;

    const int tid = threadIdx.x;

    // ---- stage reflect-padded offset neighborhoods into LDS ----
    for (int idx = tid; idx < 2 * Fc * TILE_COLS; idx += 256) {
        int ch  = idx / (Fc * TILE_COLS);
        int rem = idx - ch * (Fc * TILE_COLS);
        int r = rem / TILE_COLS;
        int c = rem - r * TILE_COLS;
        int gr = reflect_idx(h - Pc + r, Hc);
        int gc = reflect_idx(w0 - Pc + c, Wc);
        float v = (ch == 0 ? offx : offy)[gr * Wc + gc];
        (ch == 0 ? ldsX : ldsY)[r * TILE_PITCH + c] = v;
    }
    // ---- stage filter matrix (49 filters x 49 taps) ----
    for (int idx = tid; idx < Kc * Kc; idx += 256) {
        wLds[idx] = filt[idx];
    }
    __syncthreads();

    const int wave = tid >> 5;
    const int lane = tid & 31;
    const int half = lane >> 4;   // 0: K0/K1 (A) rows 0..15, 1: K2/K3
    const int lrow = lane & 15;

    const int colBase = wave * 16 + lrow;   // LDS column for this lane's pixel (halo-adjusted)

    v8f accX0 = {0,0,0,0,0,0,0,0}, accX1 = accX0, accX2 = accX0, accX3 = accX0;
    v8f accY0 = accX0, accY1 = accX0, accY2 = accX0, accY3 = accX0;

    // ---- GEMM: M=16 pixels, N=64 (49 padded), K=52 (49 padded), 13 k-steps ----
    #pragma unroll
    for (int s = 0; s < 13; ++s) {
        const int t0 = 4 * s + 2 * half;
        const int t1 = t0 + 1;

        float ax0 = 0.f, ax1 = 0.f, ay0 = 0.f, ay1 = 0.f;
        if (t0 < Kc) {
            int fi = t0 / Fc, fj = t0 - fi * Fc;
            int o = fi * TILE_PITCH + colBase + fj;
            ax0 = ldsX[o]; ay0 = ldsY[o];
        }
        if (t1 < Kc) {
            int fi = t1 / Fc, fj = t1 - fi * Fc;
            int o = fi * TILE_PITCH + colBase + fj;
            ax1 = ldsX[o]; ay1 = ldsY[o];
        }
        v2f ax = {ax0, ax1};
        v2f ay = {ay0, ay1};

        #pragma unroll
        for (int n = 0; n < 4; ++n) {
            const int k = 16 * n + lrow;
            float b0 = (k < Kc && t0 < Kc) ? wLds[k * Kc + t0] : 0.f;
            float b1 = (k < Kc && t1 < Kc) ? wLds[k * Kc + t1] : 0.f;
            v2f b = {b0, b1};
            v8f* ax_acc = (n == 0) ? &accX0 : (n == 1) ? &accX1 : (n == 2) ? &accX2 : &accX3;
            v8f* ay_acc = (n == 0) ? &accY0 : (n == 1) ? &accY1 : (n == 2) ? &accY2 : &accY3;
            *ax_acc = __builtin_amdgcn_wmma_f32_16x16x4_f32(
                false, ax, false, b, (short)0, *ax_acc, false, false);
            *ay_acc = __builtin_amdgcn_wmma_f32_16x16x4_f32(
                false, ay, false, b, (short)0, *ay_acc, false, false);
        }
    }

    // ---- bilinear sampling + accumulation over this lane's filters ----
    float sum[8][Cc];
    #pragma unroll
    for (int j = 0; j < 8; ++j)
        #pragma unroll
        for (int c = 0; c < Cc; ++c) sum[j][c] = 0.f;

    const int pwBase = w0 + wave * 16;

    #pragma unroll
    for (int n = 0; n < 4; ++n) {
        const int k = 16 * n + lrow;
        if (k >= Kc) continue;
        const v8f sx8 = (n == 0) ? accX0 : (n == 1) ? accX1 : (n == 2) ? accX2 : accX3;
        const v8f sy8 = (n == 0) ? accY0 : (n == 1) ? accY1 : (n == 2) ? accY2 : accY3;
        #pragma unroll
        for (int j = 0; j < 8; ++j) {
            const int m = j + 8 * half;   // pixel within the 16-tile (D-matrix M index)
            float cx = (float)(pwBase + m) - sx8[j];
            float cy = (float)h            - sy8[j];
            cx = fminf(fmaxf(cx, 0.f), (float)(Wc - 1));
            cy = fminf(fmaxf(cy, 0.f), (float)(Hc - 1));
            const float ix = cx - 0.5f, iy = cy - 0.5f;
            const float x0f = floorf(ix), y0f = floorf(iy);
            const float fx = ix - x0f,  fy = iy - y0f;
            const int x0 = (int)x0f, y0 = (int)y0f;
            const int x1 = x0 + 1,   y1 = y0 + 1;     // always in [0, W-1]/[0, H-1]
            const bool vx0 = (x0 >= 0), vy0 = (y0 >= 0);
            const float w00 = (1.f - fx) * (1.f - fy);
            const float w10 = fx * (1.f - fy);
            const float w01 = (1.f - fx) * fy;
            const float w11 = fx * fy;
            const int xs0 = vx0 ? x0 : 0;             // clamp for safe address
            const int ys0 = vy0 ? y0 : 0;
            #pragma unroll
            for (int c = 0; c < Cc; ++c) {
                const float* imc = img + c * (Hc * Wc);
                float v00 = imc[ys0 * Wc + xs0];
                float v10 = imc[ys0 * Wc + x1];
                float v01 = imc[y1 * Wc + xs0];
                float v11 = imc[y1 * Wc + x1];
                v00 = (vx0 && vy0) ? v00 : 0.f;
                v10 = vy0 ? v10 : 0.f;
                v01 = vx0 ? v01 : 0.f;
                sum[j][c] += w00 * v00 + w10 * v10 + w01 * v01 + w11 * v11;
            }
        }
    }

    // ---- reduce over the 16-lane half (each half owns 8 pixels, 16 lanes x ~4 filters) ----
    #pragma unroll
    for (int j = 0; j < 8; ++j) {
        #pragma unroll
        for (int c = 0; c < Cc; ++c) {
            float v = sum[j][c];
            v += __shfl_xor(v, 1, 32);
            v += __shfl_xor(v, 2, 32);
            v += __shfl_xor(v, 4, 32);
            v += __shfl_xor(v, 8, 32);
            sum[j][c] = v;
        }
    }
    if (lrow == 0) {
        const float inv = 1.0f / (float)Kc;
        #pragma unroll
        for (int j = 0; j < 8; ++j) {
            const int m = j + 8 * half;
            #pragma unroll
            for (int c = 0; c < Cc; ++c) {
                out[c * (Hc * Wc) + h * Wc + pwBase + m] = sum[j][c] * inv;
            }
        }
    }
}

extern "C" void kernel_launch(void* const* d_in, const int* in_sizes, int n_in,
                              void* d_out, int out_size, void* d_ws, size_t ws_size,
                              hipStream_t stream) {
    (void)in_sizes; (void)n_in; (void)out_size; (void)d_ws; (void)ws_size;
    const float* img  = (const float*)d_in[0];  // right_input (1,3,H,W)
    const float* offx = (const float*)d_in[1];  // offset_x (1,1,H,W)
    const float* offy = (const float*)d_in[2];  // offset_y (1,1,H,W)
    // d_in[3], d_in[4] are arange coordinate grids; recomputed in-kernel exactly.
    const float* filt = (const float*)d_in[5];  // (49,1,7,7)
    float* out = (float*)d_out;                 // (1,3,H,W)

    const int blocks = Hc * (Wc / 128);         // 3840 blocks, 8 waves each
    reconstruct_wmma_kernel<<<blocks, 256, 0, stream>>>(img, offx, offy, filt, out);
}